// GAT_6691559047386
// MI455X (gfx1250) — compile-verified
//
#include <hip/hip_runtime.h>

typedef __attribute__((ext_vector_type(2))) float v2f;
typedef __attribute__((ext_vector_type(8))) float v8f;

#define NEG_SLOPE 0.2f

// ---------------------------------------------------------------------------
// helpers: order-preserving float <-> uint key for atomicMax-based segment max
// ---------------------------------------------------------------------------
static __device__ __forceinline__ unsigned f2key(float f) {
  unsigned u = __float_as_uint(f);
  return (u & 0x80000000u) ? ~u : (u | 0x80000000u);
}
static __device__ __forceinline__ float key2f(unsigned k) {
  return (k & 0x80000000u) ? __uint_as_float(k & 0x7FFFFFFFu)
                           : __uint_as_float(~k);
}

// ---------------------------------------------------------------------------
// zero fill (grid-stride)
// ---------------------------------------------------------------------------
__global__ void zero_f32_kernel(float* __restrict__ p, long n) {
  long i = (long)blockIdx.x * blockDim.x + threadIdx.x;
  long stride = (long)gridDim.x * blockDim.x;
  for (; i < n; i += stride) p[i] = 0.0f;
}

// ---------------------------------------------------------------------------
// zero-pad copy: dst[K, ldd] <- src[K, c] (cols >= c zeroed)
// ---------------------------------------------------------------------------
__global__ void pad_cols_kernel(const float* __restrict__ src,
                                float* __restrict__ dst,
                                int K, int c, int ldd) {
  int i = blockIdx.x * blockDim.x + threadIdx.x;
  if (i >= K * ldd) return;
  int r = i / ldd, j = i - r * ldd;
  dst[i] = (j < c) ? src[r * c + j] : 0.0f;
}

// ---------------------------------------------------------------------------
// C[M, ldb] = A[M,K] * B[K, ldb]   row-major fp32 via V_WMMA_F32_16X16X4_F32.
// ldb is a multiple of 64 (B zero-padded by caller) -> inner loop is fully
// branch-free. One wave32 computes a 16(M) x 64(N) tile (4 accumulators),
// K stepped by 4, software-pipelined one step ahead: the 9 loads for step
// k+1 are issued before the 4 WMMAs for step k, so the WMMA group only
// waits on loads issued a full iteration earlier.
// ISA 16x4 f32 A layout: lane l (0..15) = row l; VGPR j, half h -> K=2h+j.
// B/C/D: row striped across lanes within a VGPR (C VGPR j -> M = j + 8*half).
// ---------------------------------------------------------------------------
__global__ void gemm_f32_wmma_kernel(const float* __restrict__ A,
                                     const float* __restrict__ B,
                                     float* __restrict__ C,
                                     int M, int K, int ldb) {
  const int lane = threadIdx.x & 31;
  const int wave = threadIdx.x >> 5;
  const int half = lane >> 4;
  const int l    = lane & 15;
  const int nt64   = ldb >> 6;
  const int mtiles = M >> 4;                 // M is a multiple of 16
  const int total  = mtiles * nt64;
  int t = blockIdx.x * (blockDim.x >> 5) + wave;
  if (t >= total) return;                    // wave-uniform: EXEC all-1s
  const int mt = t / nt64;
  const int n0 = (t - mt * nt64) << 6;
  const int m0 = mt << 4;

  v8f acc[4] = {};
  const float* arow = A + (size_t)(m0 + l) * K + 2 * half;
  const float* bp   = B + (size_t)(2 * half) * ldb + n0 + l;
  const size_t bstep = (size_t)4 * ldb;      // advance 4 K-rows per step

  // prologue: load step 0
  v2f a = *(const v2f*)(arow);
  v2f b[4];
#pragma unroll
  for (int s = 0; s < 4; ++s) {
    b[s].x = bp[s * 16];
    b[s].y = bp[ldb + s * 16];
  }

  for (int k0 = 4; k0 < K; k0 += 4) {
    // issue next step's loads first...
    v2f an = *(const v2f*)(arow + k0);
    const float* bpn = bp + bstep;
    v2f bn[4];
#pragma unroll
    for (int s = 0; s < 4; ++s) {
      bn[s].x = bpn[s * 16];
      bn[s].y = bpn[ldb + s * 16];
    }
    // ...then consume current step while next loads are in flight
#pragma unroll
    for (int s = 0; s < 4; ++s)
      acc[s] = __builtin_amdgcn_wmma_f32_16x16x4_f32(
          false, a, false, b[s], (short)0, acc[s], false, false);
    a = an;
#pragma unroll
    for (int s = 0; s < 4; ++s) b[s] = bn[s];
    bp = bpn;
  }
  // epilogue: last step
#pragma unroll
  for (int s = 0; s < 4; ++s)
    acc[s] = __builtin_amdgcn_wmma_f32_16x16x4_f32(
        false, a, false, b[s], (short)0, acc[s], false, false);

  float* crow = C + (size_t)(m0 + 8 * half) * ldb + n0 + l;
#pragma unroll
  for (int s = 0; s < 4; ++s)
#pragma unroll
    for (int j = 0; j < 8; ++j) crow[(size_t)j * ldb + s * 16] = acc[s][j];
}

// ---------------------------------------------------------------------------
// per-node attention coefficients: as[n,h] = xw[n,h,:].a_src[h,:], same for ad
// xw rows have stride ldx (>= heads*c, padded)
// ---------------------------------------------------------------------------
__global__ void alpha_kernel(const float* __restrict__ xw,
                             const float* __restrict__ a_src,
                             const float* __restrict__ a_dst,
                             float* __restrict__ as, float* __restrict__ ad,
                             int n_nodes, int heads, int c, int ldx) {
  int i = blockIdx.x * blockDim.x + threadIdx.x;
  if (i >= n_nodes * heads) return;
  int n = i / heads, h = i - n * heads;
  const float* row = xw + (size_t)n * ldx + (size_t)h * c;
  const float* ws = a_src + (size_t)h * c;
  const float* wd = a_dst + (size_t)h * c;
  float s = 0.0f, d = 0.0f;
  for (int j = 0; j < c; ++j) {
    float v = row[j];
    s += v * ws[j];
    d += v * wd[j];
  }
  as[i] = s;
  ad[i] = d;
}

// ---------------------------------------------------------------------------
// pass 1: segment max of leaky_relu(as[src]+ad[dst]) over incoming edges
// ---------------------------------------------------------------------------
__global__ void edge_max_kernel(const int* __restrict__ esrc,
                                const int* __restrict__ edst,
                                int E, int Nn, int heads,
                                const float* __restrict__ as,
                                const float* __restrict__ ad,
                                unsigned* __restrict__ mkey) {
  int e = blockIdx.x * blockDim.x + threadIdx.x;
  if (e >= E + Nn) return;
  int s = (e < E) ? esrc[e] : (e - E);   // self loops appended
  int d = (e < E) ? edst[e] : (e - E);
  for (int h = 0; h < heads; ++h) {
    float v = as[s * heads + h] + ad[d * heads + h];
    v = (v >= 0.0f) ? v : NEG_SLOPE * v;
    atomicMax(&mkey[d * heads + h], f2key(v));
  }
}

// ---------------------------------------------------------------------------
// pass 2: denom[d,h] = sum over incoming edges of exp(e - max)
// ---------------------------------------------------------------------------
__global__ void edge_denom_kernel(const int* __restrict__ esrc,
                                  const int* __restrict__ edst,
                                  int E, int Nn, int heads,
                                  const float* __restrict__ as,
                                  const float* __restrict__ ad,
                                  const unsigned* __restrict__ mkey,
                                  float* __restrict__ denom) {
  int e = blockIdx.x * blockDim.x + threadIdx.x;
  if (e >= E + Nn) return;
  int s = (e < E) ? esrc[e] : (e - E);
  int d = (e < E) ? edst[e] : (e - E);
  for (int h = 0; h < heads; ++h) {
    float v = as[s * heads + h] + ad[d * heads + h];
    v = (v >= 0.0f) ? v : NEG_SLOPE * v;
    atomicAdd(&denom[d * heads + h], expf(v - key2f(mkey[d * heads + h])));
  }
}

// ---------------------------------------------------------------------------
// pass 3: out[d,h,:] += alpha(e,h) * xw[s,h,:]   (one thread per edge x head)
// xw rows have stride ldx; out rows have stride heads*C (dense)
// ---------------------------------------------------------------------------
__global__ void edge_aggregate_kernel(const int* __restrict__ esrc,
                                      const int* __restrict__ edst,
                                      int E, int Nn, int heads, int C, int ldx,
                                      const float* __restrict__ as,
                                      const float* __restrict__ ad,
                                      const unsigned* __restrict__ mkey,
                                      const float* __restrict__ denom,
                                      const float* __restrict__ xw,
                                      float* __restrict__ out) {
  long tid = (long)blockIdx.x * blockDim.x + threadIdx.x;
  long total = (long)(E + Nn) * heads;
  if (tid >= total) return;
  int e = (int)(tid / heads);
  int h = (int)(tid - (long)e * heads);
  int s = (e < E) ? esrc[e] : (e - E);
  int d = (e < E) ? edst[e] : (e - E);
  float v = as[s * heads + h] + ad[d * heads + h];
  v = (v >= 0.0f) ? v : NEG_SLOPE * v;
  int di = d * heads + h;
  float w = expf(v - key2f(mkey[di])) / denom[di];
  const float* xs = xw + (size_t)s * ldx + (size_t)h * C;
  float* od = out + (size_t)d * heads * C + (size_t)h * C;
  for (int c = 0; c < C; ++c) atomicAdd(&od[c], w * xs[c]);
}

// ---------------------------------------------------------------------------
// epilogues
// ---------------------------------------------------------------------------
__global__ void bias_elu_kernel(float* __restrict__ h,
                                const float* __restrict__ bias, long n, int f) {
  long i = (long)blockIdx.x * blockDim.x + threadIdx.x;
  if (i >= n) return;
  float v = h[i] + bias[(int)(i % f)];
  h[i] = (v > 0.0f) ? v : (expf(v) - 1.0f);
}

__global__ void bias_add_kernel(float* __restrict__ h,
                                const float* __restrict__ bias, long n, int f) {
  long i = (long)blockIdx.x * blockDim.x + threadIdx.x;
  if (i >= n) return;
  h[i] += bias[(int)(i % f)];
}

// ---------------------------------------------------------------------------
extern "C" void kernel_launch(void* const* d_in, const int* in_sizes, int n_in,
                              void* d_out, int out_size, void* d_ws, size_t ws_size,
                              hipStream_t stream) {
  const float* x      = (const float*)d_in[0];
  const int*   ei     = (const int*)d_in[1];
  const float* W1     = (const float*)d_in[2];
  const float* a1_src = (const float*)d_in[3];
  const float* a1_dst = (const float*)d_in[4];
  const float* b1     = (const float*)d_in[5];
  const float* W2     = (const float*)d_in[6];
  const float* a2_src = (const float*)d_in[7];
  const float* a2_dst = (const float*)d_in[8];
  const float* b2     = (const float*)d_in[9];
  float* out = (float*)d_out;
  (void)n_in; (void)out_size; (void)ws_size;

  const int F_IN = 256, H1 = 8, C1 = 32, HC1 = 256, NC = 40, NCP = 64;
  const int Nn   = in_sizes[0] / F_IN;   // 50000
  const int E    = in_sizes[1] / 2;      // 800000
  const int Etot = E + Nn;
  const int* esrc = ei;
  const int* edst = ei + E;

  // workspace layout (fp32 elements); m1..agg1 and m2..den2 kept contiguous
  float* p = (float*)d_ws;
  float*    xw1  = p;            p += (size_t)Nn * HC1;
  float*    as1  = p;            p += (size_t)Nn * H1;
  float*    ad1  = p;            p += (size_t)Nn * H1;
  unsigned* m1   = (unsigned*)p; p += (size_t)Nn * H1;
  float*    den1 = p;            p += (size_t)Nn * H1;
  float*    agg1 = p;            p += (size_t)Nn * HC1;   // becomes h after ELU
  float*    w2p  = p;            p += (size_t)HC1 * NCP;  // W2 zero-padded to 64
  float*    xw2p = p;            p += (size_t)Nn * NCP;   // padded xw2
  float*    as2  = p;            p += (size_t)Nn;
  float*    ad2  = p;            p += (size_t)Nn;
  unsigned* m2   = (unsigned*)p; p += (size_t)Nn;
  float*    den2 = p;            p += (size_t)Nn;

  dim3 blk(256);

  // zero: [m1 | den1 | agg1] contiguous, [m2 | den2] contiguous, and d_out
  zero_f32_kernel<<<2048, blk, 0, stream>>>((float*)m1,
                                            (long)Nn * (2 * H1 + HC1));
  zero_f32_kernel<<<256, blk, 0, stream>>>((float*)m2, (long)Nn * 2);
  zero_f32_kernel<<<1024, blk, 0, stream>>>(out, (long)Nn * NC);
  // pad W2 [256,40] -> [256,64]
  pad_cols_kernel<<<(HC1 * NCP + 255) / 256, blk, 0, stream>>>(
      W2, w2p, HC1, NC, NCP);

  // ---- layer 1 ----
  {
    int waves  = (Nn / 16) * (HC1 / 64);
    int blocks = (waves + 7) / 8;
    gemm_f32_wmma_kernel<<<blocks, blk, 0, stream>>>(x, W1, xw1, Nn, F_IN, HC1);
  }
  alpha_kernel<<<(Nn * H1 + 255) / 256, blk, 0, stream>>>(
      xw1, a1_src, a1_dst, as1, ad1, Nn, H1, C1, HC1);
  edge_max_kernel<<<(Etot + 255) / 256, blk, 0, stream>>>(
      esrc, edst, E, Nn, H1, as1, ad1, m1);
  edge_denom_kernel<<<(Etot + 255) / 256, blk, 0, stream>>>(
      esrc, edst, E, Nn, H1, as1, ad1, m1, den1);
  {
    long tot = (long)Etot * H1;
    edge_aggregate_kernel<<<(int)((tot + 255) / 256), blk, 0, stream>>>(
        esrc, edst, E, Nn, H1, C1, HC1, as1, ad1, m1, den1, xw1, agg1);
  }
  bias_elu_kernel<<<(int)(((long)Nn * HC1 + 255) / 256), blk, 0, stream>>>(
      agg1, b1, (long)Nn * HC1, HC1);

  // ---- layer 2 ----
  {
    int waves  = (Nn / 16) * (NCP / 64);
    int blocks = (waves + 7) / 8;
    gemm_f32_wmma_kernel<<<blocks, blk, 0, stream>>>(agg1, w2p, xw2p,
                                                     Nn, HC1, NCP);
  }
  alpha_kernel<<<(Nn + 255) / 256, blk, 0, stream>>>(
      xw2p, a2_src, a2_dst, as2, ad2, Nn, 1, NC, NCP);
  edge_max_kernel<<<(Etot + 255) / 256, blk, 0, stream>>>(
      esrc, edst, E, Nn, 1, as2, ad2, m2);
  edge_denom_kernel<<<(Etot + 255) / 256, blk, 0, stream>>>(
      esrc, edst, E, Nn, 1, as2, ad2, m2, den2);
  edge_aggregate_kernel<<<(Etot + 255) / 256, blk, 0, stream>>>(
      esrc, edst, E, Nn, 1, NC, NCP, as2, ad2, m2, den2, xw2p, out);
  bias_add_kernel<<<(int)(((long)Nn * NC + 255) / 256), blk, 0, stream>>>(
      out, b2, (long)Nn * NC, NC);
}